// ROIPooling3D_41996190220506
// MI455X (gfx1250) — compile-verified
//
#include <hip/hip_runtime.h>
#include <stdint.h>

// Problem constants (from reference)
#define NB   4
#define NC   256
#define NVOX (46*55*46)        // 116380, divisible by 4
#define NROI 200

// Tiling
#define CT        64           // channels per tile
#define NCT       (NC/CT)      // 4
#define CHUNK     1024         // voxels per TDM chunk (4 KB of int32, x2 buffers)
#define NCHUNKS   ((NVOX + CHUNK - 1)/CHUNK)   // 114
#define BLKS_PER_PAIR 32
#define THREADS   256

typedef uint32_t u32x4 __attribute__((ext_vector_type(4)));
typedef int32_t  i32x8 __attribute__((ext_vector_type(8)));
typedef int32_t  i32x4 __attribute__((ext_vector_type(4)));

// 1-D TDM copy (async): global int32[] -> LDS, tile length `tile_len`; OOB
// reads beyond `remaining` elements return 0.  D# per CDNA5 ISA ch.8.
// Caller must s_wait_tensorcnt before consuming the LDS data.
__device__ __forceinline__ void tdm_issue_1d_to_lds(uint32_t lds_off, uint64_t gaddr,
                                                    uint32_t remaining, uint32_t tile_len) {
  u32x4 g0;
  g0.x = 1u;                                                  // count=1 (valid user D#)
  g0.y = lds_off;                                             // lds_addr (bytes)
  g0.z = (uint32_t)gaddr;                                     // global_addr[31:0]
  g0.w = (uint32_t)((gaddr >> 32) & 0x1FFFFFFu) | (2u << 30); // addr[56:32] | type=2

  i32x8 g1;
  g1[0] = (int)(2u << 16);                                    // data_size = 4 bytes
  g1[1] = (int)((remaining & 0xFFFFu) << 16);                 // tensor_dim0[15:0]
  g1[2] = (int)(((remaining >> 16) & 0xFFFFu) | (1u << 16));  // tensor_dim0[31:16] | tensor_dim1=1
  g1[3] = (int)((tile_len & 0xFFFFu) << 16);                  // tile_dim0
  g1[4] = 0;                                                  // tile_dim1=0 (1-D), tile_dim2=0
  g1[5] = (int)remaining;                                     // tensor_dim0_stride lo
  g1[6] = 0;
  g1[7] = 0;

  i32x4 gz  = {0, 0, 0, 0};                                   // dims 2..4 unused
  i32x8 gz8 = {0, 0, 0, 0, 0, 0, 0, 0};
  __builtin_amdgcn_tensor_load_to_lds(g0, g1, gz, gz, gz8, 0);
}

__global__ void __launch_bounds__(256) zero_kernel(float* p, int n) {
  int i = blockIdx.x * 256 + threadIdx.x;
  if (i < n) p[i] = 0.0f;
}

// Per-ROI voxel counts via LDS histogram, then one global atomic per bin.
__global__ void __launch_bounds__(256) counts_kernel(const int* __restrict__ atlas,
                                                     float* __restrict__ counts) {
  __shared__ float h[NROI];
  for (int i = threadIdx.x; i < NROI; i += 256) h[i] = 0.0f;
  __syncthreads();
  for (int i = blockIdx.x * 256 + threadIdx.x; i < NVOX; i += gridDim.x * 256) {
    int s = atlas[i];
    if (s > 0) atomicAdd(&h[s - 1], 1.0f);
  }
  __syncthreads();
  for (int i = threadIdx.x; i < NROI; i += 256) {
    float v = h[i];
    if (v != 0.0f) atomicAdd(&counts[i], v);
  }
}

// Main streaming segment-sum: double-buffered TDM atlas chunks -> LDS,
// coalesced float4 feature stream, ds_add_f32 accumulation into LDS
// [chan][roi] (per-lane bank = f(roi), random spread), single global flush.
__global__ void __launch_bounds__(THREADS) seg_sum_kernel(const float* __restrict__ feat,
                                                          const int* __restrict__ atlas,
                                                          float* __restrict__ sums) {
  __shared__ float accum[CT * NROI];   // 51200 B
  __shared__ int   seg[2][CHUNK];      //  8192 B (ping-pong)

  const int pair = blockIdx.x / BLKS_PER_PAIR;   // 0..15 -> (channel tile, batch)
  const int sub  = blockIdx.x % BLKS_PER_PAIR;
  const int ct   = pair % NCT;
  const int b    = pair / NCT;
  const int c0   = ct * CT;
  const int tid  = threadIdx.x;
  const int wave = tid >> 5;
  const int lane = tid & 31;

  for (int i = tid; i < CT * NROI; i += THREADS) accum[i] = 0.0f;

  // Prologue: kick off the DMA for this block's first chunk into buffer 0.
  if (tid < 32 && sub < NCHUNKS) {
    const int vbase0 = sub * CHUNK;
    tdm_issue_1d_to_lds((uint32_t)(uintptr_t)(void*)seg[0],
                        (uint64_t)(uintptr_t)(const void*)(atlas + vbase0),
                        (uint32_t)(NVOX - vbase0), CHUNK);
  }

  int p = 0;
  for (int chunk = sub; chunk < NCHUNKS; chunk += BLKS_PER_PAIR) {
    const int vbase = chunk * CHUNK;
    const int vlen  = (NVOX - vbase) < CHUNK ? (NVOX - vbase) : CHUNK;

    if (tid < 32) __builtin_amdgcn_s_wait_tensorcnt(0);   // chunk data landed
    __syncthreads();                                      // all waves: seg[p] ready,
                                                          // seg[1-p] reads finished
    const int nextChunk = chunk + BLKS_PER_PAIR;
    if (tid < 32 && nextChunk < NCHUNKS) {                // overlap next DMA w/ compute
      const int nvbase = nextChunk * CHUNK;
      tdm_issue_1d_to_lds((uint32_t)(uintptr_t)(void*)seg[1 - p],
                          (uint64_t)(uintptr_t)(const void*)(atlas + nvbase),
                          (uint32_t)(NVOX - nvbase), CHUNK);
    }

    // 8 waves x 8 channel passes = 64 channels; lanes stream contiguous voxels.
    const int* __restrict__ sp = seg[p];
    for (int coff = wave; coff < CT; coff += (THREADS / 32)) {
      const float4* __restrict__ fp =
          (const float4*)(feat + ((size_t)(b * NC + c0 + coff)) * (size_t)NVOX + vbase);
      float* __restrict__ acc = &accum[coff * NROI];
      const int n4 = vlen >> 2;            // vlen always divisible by 4
      for (int i4 = lane; i4 < n4; i4 += 32) {
        float4 f = fp[i4];
        int vb = i4 << 2;
        int s0 = sp[vb + 0]; if (s0 > 0) atomicAdd(&acc[s0 - 1], f.x);
        int s1 = sp[vb + 1]; if (s1 > 0) atomicAdd(&acc[s1 - 1], f.y);
        int s2 = sp[vb + 2]; if (s2 > 0) atomicAdd(&acc[s2 - 1], f.z);
        int s3 = sp[vb + 3]; if (s3 > 0) atomicAdd(&acc[s3 - 1], f.w);
      }
    }
    __syncthreads();                       // done reading seg[p] before next overwrite
    p ^= 1;
  }
  __syncthreads();

  // Flush block-partial sums: sums layout [B][NROI][NC]
  for (int i = tid; i < CT * NROI; i += THREADS) {
    float v = accum[i];
    if (v != 0.0f) {
      int coff = i / NROI;
      int r    = i % NROI;
      atomicAdd(&sums[((size_t)b * NROI + r) * NC + (c0 + coff)], v);
    }
  }
}

// out[0 .. B*NROI*NC)          = roi_features (mean or 0)
// out[B*NROI*NC .. +B*NROI)    = validity mask as 1.0/0.0
__global__ void __launch_bounds__(256) finalize_kernel(const float* __restrict__ sums,
                                                       const float* __restrict__ counts,
                                                       float* __restrict__ out) {
  const int nfeat = NB * NROI * NC;
  int i = blockIdx.x * 256 + threadIdx.x;
  if (i < nfeat) {
    int r = (i / NC) % NROI;
    float c = counts[r];
    out[i] = (c > 0.0f) ? (sums[i] / c) : 0.0f;
  } else if (i < nfeat + NB * NROI) {
    int r = (i - nfeat) % NROI;
    out[i] = (counts[r] > 0.0f) ? 1.0f : 0.0f;
  }
}

extern "C" void kernel_launch(void* const* d_in, const int* in_sizes, int n_in,
                              void* d_out, int out_size, void* d_ws, size_t ws_size,
                              hipStream_t stream) {
  const float* feat  = (const float*)d_in[0];
  const int*   atlas = (const int*)d_in[1];
  float* out    = (float*)d_out;
  float* sums   = (float*)d_ws;                      // NB*NROI*NC floats
  float* counts = sums + (size_t)NB * NROI * NC;     // NROI floats

  const int zn = NB * NROI * NC + NROI;
  zero_kernel<<<(zn + 255) / 256, 256, 0, stream>>>(sums, zn);
  counts_kernel<<<128, 256, 0, stream>>>(atlas, counts);
  seg_sum_kernel<<<NCT * NB * BLKS_PER_PAIR, THREADS, 0, stream>>>(feat, atlas, sums);

  const int total = NB * NROI * NC + NB * NROI;
  finalize_kernel<<<(total + 255) / 256, 256, 0, stream>>>(sums, counts, out);
}